// InfoNCELoss_33122787787557
// MI455X (gfx1250) — compile-verified
//
#include <hip/hip_runtime.h>
#include <hip/hip_bf16.h>
#include <stdint.h>

// Problem constants (from reference: B=192, T=100, V=4096)
#define B_      192
#define T_      100
#define V_      4096
#define NC_     10          // number of t-chunks
#define CHUNK_  (T_ / NC_)  // 10 rows per chunk
#define BLK_    256         // 8 wave32 waves
#define NCOLS_  (B_ * B_)   // 36864 flattened (j,i) columns of the partials

typedef __attribute__((ext_vector_type(2))) float v2f;
typedef __attribute__((ext_vector_type(8))) float v8f;

// Issue one 16KB row (4096 f32) as async global->LDS DMA:
// 4 x global_load_async_to_lds_b128 per thread (256 thr * 16B * 4 = 16KB).
// GVS addressing: mem = SGPR_base + VGPR_off; LDS dest = per-lane VGPR addr.
__device__ __forceinline__ void async_row_load(const float* __restrict__ xbase,
                                               unsigned ldsBase, unsigned rowByte,
                                               unsigned tid) {
#pragma unroll
  for (int k = 0; k < 4; ++k) {
    unsigned off = tid * 16u + (unsigned)k * 4096u;  // bytes within the row
    asm volatile("global_load_async_to_lds_b128 %0, %1, %2"
                 :
                 : "v"(ldsBase + off), "v"(rowByte + off), "s"(xbase)
                 : "memory");
  }
}

// Kernel 1: block (j, c) accumulates partial column P[c][j][i] =
//   sum_{t in chunk c} clip(x[j,t,targets[i,t]])
__global__ __launch_bounds__(BLK_) void
infonce_partial_kernel(const float* __restrict__ x,
                       const int* __restrict__ targets,
                       float* __restrict__ P) {
  __shared__ float smem[2][V_];  // 2 x 16KB double buffer
  const int j   = blockIdx.x;
  const int c   = blockIdx.y;
  const int c0  = c * CHUNK_;
  const int tid = threadIdx.x;

  // Preload this thread's gather indices for the chunk (kept in registers).
  int tg[CHUNK_];
  if (tid < B_) {
#pragma unroll
    for (int tt = 0; tt < CHUNK_; ++tt)
      tg[tt] = targets[tid * T_ + c0 + tt];
  }

  // LDS byte offsets of the two buffers (flat-addr low 32 bits == LDS offset).
  const unsigned lds0 = (unsigned)(uintptr_t)&smem[0][0];
  const unsigned lds1 = (unsigned)(uintptr_t)&smem[1][0];

  // Prefetch first row of the chunk.
  {
    unsigned rowByte = (unsigned)((((size_t)j * T_ + c0) * V_) * 4u);
    async_row_load(x, lds0, rowByte, tid);
  }

  float acc = 0.f;
#pragma unroll
  for (int tt = 0; tt < CHUNK_; ++tt) {
    // Wait for this wave's outstanding async DMAs (row tt), then make the
    // full row visible workgroup-wide. The barrier also guarantees everyone
    // is done gathering from the buffer we are about to overwrite.
    asm volatile("s_wait_asynccnt 0x0" ::: "memory");
    __syncthreads();

    // Kick off next row's DMA into the other buffer: overlaps with gathers.
    if (tt + 1 < CHUNK_) {
      unsigned rowByte =
          (unsigned)((((size_t)j * T_ + (c0 + tt + 1)) * V_) * 4u);
      async_row_load(x, ((tt + 1) & 1) ? lds1 : lds0, rowByte, tid);
    }

    // 192 LDS gathers + clip + accumulate.
    if (tid < B_) {
      float v = smem[tt & 1][tg[tt]];
      acc += fminf(fmaxf(v, -30.f), 30.f);
    }
  }

  // Coalesced partial store: P[c][j][i], i fastest.
  if (tid < B_)
    P[((size_t)c * B_ + j) * B_ + tid] = acc;
}

// Kernel 2: Sbuf[n] = sum_{c=0..NC_-1} P[c][n] via the matrix pipe.
// Colsum == ones(16x4) x P_tile(4x16) accumulated over 3 chained WMMAs
// (K = 4+4+2, zero-padded). With A == 1 the result is invariant to the
// hardware's K-slot permutation in the B layout: each lane loads 2 distinct
// chunk values per WMMA (slot s = (laneHi)*2 + vgpr), zero for c >= NC_.
// Each wave produces 16 columns; 36864/16 = 2304 waves, no remainder, so
// every wave runs WMMA with EXEC all-ones and no divergence.
__global__ __launch_bounds__(BLK_) void
infonce_wmma_reduce_kernel(const float* __restrict__ P,
                           float* __restrict__ Sbuf) {
  const int lane  = threadIdx.x & 31;
  const int wave  = (blockIdx.x * (BLK_ / 32)) + (threadIdx.x >> 5);
  const int n0    = wave * 16;
  const int n     = n0 + (lane & 15);
  const int sHalf = (lane >> 4) & 1;  // which half-wave this lane is in

  // All-ones A (16x4 f32 -> 2 VGPRs per lane).
  v2f a;
  a.x = 1.0f;
  a.y = 1.0f;

  v8f acc = {};  // C = 0 for the first WMMA, then chained
#pragma unroll
  for (int g = 0; g < 3; ++g) {
    v2f b;
    {
      int c0 = g * 4 + sHalf * 2 + 0;  // slot for B vgpr 0
      int c1 = g * 4 + sHalf * 2 + 1;  // slot for B vgpr 1
      b.x = (c0 < NC_) ? P[(size_t)c0 * NCOLS_ + n] : 0.0f;
      b.y = (c1 < NC_) ? P[(size_t)c1 * NCOLS_ + n] : 0.0f;
    }
    acc = __builtin_amdgcn_wmma_f32_16x16x4_f32(
        /*neg_a=*/false, a, /*neg_b=*/false, b,
        /*c_mod=*/(short)0, acc, /*reuse_a=*/false, /*reuse_b=*/false);
  }

  // D row M=0 lives in VGPR0, lanes 0-15 (N = lane). All rows of D are
  // identical (rows of A are equal), so this is layout-robust.
  if (lane < 16)
    Sbuf[n0 + lane] = acc.s0;
}

// Kernel 3: per-row logsumexp + diagonal, tree-reduce to scalar loss.
// Thread i owns S[i, :] = Sbuf[j*B + i] for all j.
__global__ __launch_bounds__(256) void
infonce_finalize_kernel(const float* __restrict__ Sbuf,
                        float* __restrict__ out) {
  const int i = threadIdx.x;
  float r = 0.f;
  if (i < B_) {
    float m = -3.0e38f;
#pragma unroll 1
    for (int jj = 0; jj < B_; ++jj)
      m = fmaxf(m, Sbuf[jj * B_ + i]);  // lanes contiguous in i: coalesced
    float sum = 0.f;
#pragma unroll 1
    for (int jj = 0; jj < B_; ++jj)
      sum += expf(Sbuf[jj * B_ + i] - m);
    float lse   = m + logf(sum);
    float numer = Sbuf[i * B_ + i];  // S[i,i]
    r = numer - lse;
  }

  __shared__ float red[256];
  red[i] = r;
  __syncthreads();
#pragma unroll
  for (int s = 128; s > 0; s >>= 1) {
    if (i < s) red[i] += red[i + s];
    __syncthreads();
  }
  if (i == 0)
    out[0] = -red[0] / (float)(B_ * T_);
}

extern "C" void kernel_launch(void* const* d_in, const int* in_sizes, int n_in,
                              void* d_out, int out_size, void* d_ws, size_t ws_size,
                              hipStream_t stream) {
  const float* x       = (const float*)d_in[0];
  const int*   targets = (const int*)d_in[1];

  // Workspace layout: P[NC][B][B] partials, then Sbuf[B*B]. ~1.62 MB total.
  float* P    = (float*)d_ws;
  float* Sbuf = P + (size_t)NC_ * NCOLS_;
  float* out  = (float*)d_out;

  dim3 grid1(B_, NC_);
  infonce_partial_kernel<<<grid1, BLK_, 0, stream>>>(x, targets, P);

  // 2304 waves of 16 columns each; 8 waves/block -> 288 blocks.
  infonce_wmma_reduce_kernel<<<NCOLS_ / 16 / (BLK_ / 32), BLK_, 0, stream>>>(P, Sbuf);

  infonce_finalize_kernel<<<1, 256, 0, stream>>>(Sbuf, out);
}